// WSGConv_17600775979419
// MI455X (gfx1250) — compile-verified
//
#include <hip/hip_runtime.h>
#include <math.h>

typedef __attribute__((ext_vector_type(2))) float v2f;
typedef __attribute__((ext_vector_type(8))) float v8f;

#define D 64
#define K3D 192

// ---------------- kernel 1: zero workspace ----------------
__global__ void k_zero(float* ws, long n) {
    long i = (long)blockIdx.x * blockDim.x + threadIdx.x;
    long stride = (long)gridDim.x * blockDim.x;
    for (; i < n; i += stride) ws[i] = 0.0f;
}

// ---------------- kernel 2: segment max (uint trick on positive floats) ----
__global__ void k_segmax(const float* __restrict__ w, const int* __restrict__ dst,
                         unsigned int* __restrict__ m_pos, unsigned int* __restrict__ m_neg,
                         int E) {
    int e = blockIdx.x * blockDim.x + threadIdx.x;
    if (e >= E) return;
    float wv = w[e];
    int d = dst[e];
    if (wv > 0.0f) {
        atomicMax(&m_pos[d], __float_as_uint(wv));
    } else if (wv < 0.0f) {
        atomicMax(&m_neg[d], __float_as_uint(-wv));
    }
}

// ---------------- kernel 3: exp + segment sum ----------------
__global__ void k_expsum(const float* __restrict__ w, const int* __restrict__ dst,
                         const float* __restrict__ m_pos, const float* __restrict__ m_neg,
                         float* __restrict__ e_pos, float* __restrict__ e_neg,
                         float* __restrict__ s_pos, float* __restrict__ s_neg,
                         int E) {
    int e = blockIdx.x * blockDim.x + threadIdx.x;
    if (e >= E) return;
    float wv = w[e];
    int d = dst[e];
    float ep = 0.0f, en = 0.0f;
    if (wv > 0.0f) {
        ep = expf(wv - m_pos[d]);
        atomicAdd(&s_pos[d], ep);
    } else if (wv < 0.0f) {
        en = expf(-wv - m_neg[d]);
        atomicAdd(&s_neg[d], en);
    }
    e_pos[e] = ep;
    e_neg[e] = en;
}

// ---------------- kernel 4: weighted gather/scatter (one thread per edge,dim)
__global__ void k_scatter(const float* __restrict__ feat,
                          const float* __restrict__ w,
                          const int* __restrict__ src, const int* __restrict__ dst,
                          const float* __restrict__ e_pos, const float* __restrict__ e_neg,
                          const float* __restrict__ s_pos, const float* __restrict__ s_neg,
                          float* __restrict__ h_pos, float* __restrict__ h_neg,
                          int E) {
    int t = blockIdx.x * blockDim.x + threadIdx.x;
    int e = t >> 6;         // edge
    int d = t & 63;         // feature dim
    if (e >= E) return;
    float wv = w[e];
    if (wv == 0.0f) return;
    int s = src[e];
    int dd = dst[e];
    float f = feat[s * D + d];
    if (wv > 0.0f) {
        float a = e_pos[e] / (s_pos[dd] + 1e-16f);
        atomicAdd(&h_pos[dd * D + d], a * f);
    } else {
        float a = e_neg[e] / (s_neg[dd] + 1e-16f);
        atomicAdd(&h_neg[dd * D + d], a * f);
    }
}

// ---------------- kernel 5: fp32 WMMA GEMM  out = X @ W^T + b_fc + bias ----
// X[r][k] = concat(cs*feat, cp*h_pos, cn*h_neg)[r][k],  W = W_fc [64 x 192]
// Block: 128 threads = 4 waves; each wave owns one 16-col tile of a 16-row strip.
__device__ __forceinline__ float xval(const float* feat, const float* hp, const float* hn,
                                      float cs, float cp, float cn, int r, int k) {
    if (k < D)       return cs * feat[r * D + k];
    if (k < 2 * D)   return cp * hp[r * D + (k - D)];
    return cn * hn[r * D + (k - 2 * D)];
}

__global__ void k_gemm_wmma(const float* __restrict__ feat,
                            const float* __restrict__ h_pos,
                            const float* __restrict__ h_neg,
                            const float* __restrict__ Wfc,   // [64,192] row-major
                            const float* __restrict__ bfc,   // [64]
                            const float* __restrict__ bias,  // [64]
                            const float* __restrict__ cs_p,
                            const float* __restrict__ cp_p,
                            const float* __restrict__ cn_p,
                            float* __restrict__ out,         // [N,64]
                            int nrows) {
    const int lane   = threadIdx.x & 31;
    const int wave   = threadIdx.x >> 5;     // 0..3 -> output column tile
    const int lo16   = lane & 15;
    const int hi     = lane >> 4;            // 0 or 1
    const int r      = blockIdx.x * 16 + lo16;     // A-matrix row for this lane
    const int kb     = hi * 2;                     // A/B K sub-offset (0 or 2)
    const int ncol   = wave * 16 + lo16;           // B-matrix column for this lane

    const float cs = cs_p[0], cp = cp_p[0], cn = cn_p[0];

    v8f acc = {};
    #pragma unroll
    for (int k0 = 0; k0 < K3D; k0 += 4) {
        // A tile 16x4 fp32: lane holds (M=lane%16, K=k0+kb .. k0+kb+1)
        v2f a;
        a.x = xval(feat, h_pos, h_neg, cs, cp, cn, r, k0 + kb);
        a.y = xval(feat, h_pos, h_neg, cs, cp, cn, r, k0 + kb + 1);
        // B tile 4x16 fp32: lane holds (N=lane%16, K=k0+kb .. k0+kb+1); B[k][n] = W[n][k]
        v2f b;
        b.x = Wfc[ncol * K3D + k0 + kb];
        b.y = Wfc[ncol * K3D + k0 + kb + 1];
        acc = __builtin_amdgcn_wmma_f32_16x16x4_f32(
            /*neg_a=*/false, a, /*neg_b=*/false, b,
            /*c_mod=*/(short)0, acc, /*reuse_a=*/false, /*reuse_b=*/false);
    }

    // C/D layout: VGPR j of lane L holds (M = j + 8*(L/16), N = L%16)
    #pragma unroll
    for (int j = 0; j < 8; ++j) {
        int m   = j + 8 * hi;
        int row = blockIdx.x * 16 + m;
        if (row < nrows) {
            int col = wave * 16 + lo16;
            out[row * D + col] = acc[j] + bfc[col] + bias[col];
        }
    }
}

extern "C" void kernel_launch(void* const* d_in, const int* in_sizes, int n_in,
                              void* d_out, int out_size, void* d_ws, size_t ws_size,
                              hipStream_t stream) {
    const float* feat = (const float*)d_in[0];   // [N,64]
    const float* w    = (const float*)d_in[1];   // [E]
    const float* Wfc  = (const float*)d_in[2];   // [64,192]
    const float* bfc  = (const float*)d_in[3];   // [64]
    const float* bias = (const float*)d_in[4];   // [64]
    const float* cs   = (const float*)d_in[5];
    const float* cp   = (const float*)d_in[6];
    const float* cn   = (const float*)d_in[7];
    const int*   src  = (const int*)d_in[8];     // [E]
    const int*   dst  = (const int*)d_in[9];     // [E]
    float*       out  = (float*)d_out;

    const int N = in_sizes[0] / D;
    const int E = in_sizes[1];

    // workspace layout (floats)
    float* ws    = (float*)d_ws;
    float* m_pos = ws;                    // N   (float bits; max'd as uint)
    float* m_neg = m_pos + N;             // N
    float* s_pos = m_neg + N;             // N
    float* s_neg = s_pos + N;             // N
    float* e_pos = s_neg + N;             // E
    float* e_neg = e_pos + E;             // E
    float* h_pos = e_neg + E;             // N*D
    float* h_neg = h_pos + (long)N * D;   // N*D
    long   ws_floats = 4L * N + 2L * E + 2L * N * D;

    // 1) zero workspace
    k_zero<<<2048, 256, 0, stream>>>(ws, ws_floats);

    // 2) segment max per dst (pos: w, neg: |w|) via uint atomicMax
    int eb = (E + 255) / 256;
    k_segmax<<<eb, 256, 0, stream>>>(w, dst, (unsigned int*)m_pos, (unsigned int*)m_neg, E);

    // 3) exp(l - m[dst]) and segment sums
    k_expsum<<<eb, 256, 0, stream>>>(w, dst, m_pos, m_neg, e_pos, e_neg, s_pos, s_neg, E);

    // 4) normalized gather/scatter into h_pos / h_neg
    long t4 = (long)E * D;
    int  b4 = (int)((t4 + 255) / 256);
    k_scatter<<<b4, 256, 0, stream>>>(feat, w, src, dst, e_pos, e_neg, s_pos, s_neg,
                                      h_pos, h_neg, E);

    // 5) fp32 WMMA GEMM: [N,192] x [192,64]^T + b_fc + bias
    int rb = (N + 15) / 16;
    k_gemm_wmma<<<rb, 128, 0, stream>>>(feat, h_pos, h_neg, Wfc, bfc, bias,
                                        cs, cp, cn, out, N);
}